// GConvSE3Partial_48902497632448
// MI455X (gfx1250) — compile-verified
//
#include <hip/hip_runtime.h>

typedef __attribute__((ext_vector_type(2))) float v2f;
typedef __attribute__((ext_vector_type(8))) float v8f;

#define E_EDGES   160000
#define EDGE_DIMC 16
#define MIDC      32
#define LN_EPS    1e-5f
#define WAVES     4

struct RadialParams {
  const float* W1; const float* b1; const float* g1; const float* be1;
  const float* W2; const float* b2; const float* g2; const float* be2;
  const float* W3; const float* b3;
};
struct AllParams { RadialParams p[4]; };

struct WaveLDS {
  float x[16 * 36];   // normalized activations, row stride 36 (conflict-free)
  float R[16 * 100];  // layer-3 section outputs; also reused to stage feat [16][20]
  float basis[544];   // b00[16] | b01[48] | b10[48] | b11[432]
};

__device__ __forceinline__ v8f wmma4(v2f a, v2f b, v8f c) {
  // V_WMMA_F32_16X16X4_F32: exact fp32 MACs on the matrix pipe
  return __builtin_amdgcn_wmma_f32_16x16x4_f32(false, a, false, b, (short)0, c,
                                               false, false);
}

__device__ __forceinline__ void ln_relu_store(float* xbuf, v8f acc0, v8f acc1,
                                              const float* bias, const float* g,
                                              const float* be, int half, int mm) {
  const float biasA = bias[mm],  biasB = bias[16 + mm];
  const float gA = g[mm],        gB = g[16 + mm];
  const float beA = be[mm],      beB = be[16 + mm];
#pragma unroll
  for (int v = 0; v < 8; ++v) {
    float x0 = acc0[v] + biasA;
    float x1 = acc1[v] + biasB;
    float s = x0 + x1;
    float q = x0 * x0 + x1 * x1;
#pragma unroll
    for (int m = 1; m < 16; m <<= 1) {  // stays within 16-lane half = one matrix row
      s += __shfl_xor(s, m, 32);
      q += __shfl_xor(q, m, 32);
    }
    float mean = s * (1.f / 32.f);
    float var  = q * (1.f / 32.f) - mean * mean;
    float inv  = rsqrtf(var + LN_EPS);
    float y0 = fmaxf((x0 - mean) * inv * gA + beA, 0.f);
    float y1 = fmaxf((x1 - mean) * inv * gB + beB, 0.f);
    const int row = v + 8 * half;
    xbuf[row * 36 + mm]      = y0;
    xbuf[row * 36 + 16 + mm] = y1;
  }
}

__global__ __launch_bounds__(WAVES * 32, 1) void se3_partial_wmma(
    const float* __restrict__ h0, const float* __restrict__ h1,
    const float* __restrict__ r, const float* __restrict__ edge_feat,
    const int* __restrict__ edge_row,
    const float* __restrict__ basis00, const float* __restrict__ basis01,
    const float* __restrict__ basis10, const float* __restrict__ basis11,
    AllParams params, float* __restrict__ out) {
  __shared__ WaveLDS lds[WAVES];
  const int lane = threadIdx.x & 31;
  const int wave = threadIdx.x >> 5;
  const int e0 = (blockIdx.x * WAVES + wave) * 16;
  const int half = lane >> 4;
  const int mm = lane & 15;
  WaveLDS& L = lds[wave];

  // ---- stage edge features (coalesced float4) into L.R used as [16][20] ----
  {
    const int fe = lane >> 1, part = lane & 1;
    const float4* src =
        (const float4*)(edge_feat + (size_t)(e0 + fe) * EDGE_DIMC + part * 8);
    float4 f0 = src[0], f1 = src[1];
    float* dst = &L.R[fe * 20 + part * 8];
    dst[0] = f0.x; dst[1] = f0.y; dst[2] = f0.z; dst[3] = f0.w;
    dst[4] = f1.x; dst[5] = f1.y; dst[6] = f1.z; dst[7] = f1.w;
    if (part == 0) {
      L.R[fe * 20 + 16] = r[e0 + fe];            // feat = [edge_feat | r]
    } else {
      L.R[fe * 20 + 17] = 0.f; L.R[fe * 20 + 18] = 0.f; L.R[fe * 20 + 19] = 0.f;
    }
  }
  // ---- stage basis tiles (coalesced) ----
  for (int i = lane; i < 16;  i += 32) L.basis[i]       = basis00[e0 + i];
  for (int i = lane; i < 48;  i += 32) L.basis[16 + i]  = basis01[(size_t)e0 * 3 + i];
  for (int i = lane; i < 48;  i += 32) L.basis[64 + i]  = basis10[(size_t)e0 * 3 + i];
  for (int i = lane; i < 432; i += 32) L.basis[112 + i] = basis11[(size_t)e0 * 27 + i];
  __builtin_amdgcn_wave_barrier();

  // ---- A fragments, layer 1 (K padded 17 -> 20 with staged zeros) ----
  v2f a1[5];
#pragma unroll
  for (int c = 0; c < 5; ++c) {
    const int k = 4 * c + 2 * half;
    a1[c].x = L.R[mm * 20 + k];
    a1[c].y = L.R[mm * 20 + k + 1];
  }

  // ---- per-edge epilogue operands: 2 lanes per edge ----
  const int me = lane >> 1;
  const int sub = lane & 1;
  const int eg = e0 + me;
  const int rown = edge_row[eg];          // edge_index[0][e]
  float s0[8], s1[8][3], t10[8];
#pragma unroll
  for (int b = 0; b < 8; ++b) s0[b] = h0[(size_t)rown * 8 + b];
#pragma unroll
  for (int b = 0; b < 8; ++b)
#pragma unroll
    for (int i = 0; i < 3; ++i) s1[b][i] = h1[(size_t)rown * 24 + b * 3 + i];
  const float b00v = L.basis[me];
  float b01v[3], b10v[3];
#pragma unroll
  for (int j = 0; j < 3; ++j) {
    b01v[j] = L.basis[16 + me * 3 + j];
    b10v[j] = L.basis[64 + me * 3 + j];
  }
#pragma unroll
  for (int b = 0; b < 8; ++b)
    t10[b] = b10v[0] * s1[b][0] + b10v[1] * s1[b][1] + b10v[2] * s1[b][2];

  float out0r[4] = {0.f, 0.f, 0.f, 0.f};
  float out1r[4][3] = {};
  const v8f Z8 = {0.f, 0.f, 0.f, 0.f, 0.f, 0.f, 0.f, 0.f};

  for (int p = 0; p < 4; ++p) {
    const RadialParams P = params.p[p];
    const int outdim = (p == 3) ? 192 : 64;

    // Warm WGP$ for this pair's weight streams (global_prefetch_b8 path).
    __builtin_prefetch(P.W1, 0, 1);
    __builtin_prefetch(P.W2, 0, 1);
    __builtin_prefetch(P.W3, 0, 1);

    // ---- layer 1: [16,17] x [17,32] (K padded to 20) ----
    // Branch-free tail handling: clamp the row index to a valid row and
    // zero the *value* with a select (v_cndmask), so no exec-masked loads.
    v8f acc0 = Z8, acc1 = Z8;
#pragma unroll
    for (int c = 0; c < 5; ++c) {
      const int k0 = 4 * c + 2 * half;
      const int ra = (k0     < 17) ? k0     : 16;   // clamped, always in-bounds
      const int rb = (k0 + 1 < 17) ? k0 + 1 : 16;
      const float wa0 = P.W1[ra * MIDC + mm];
      const float wa1 = P.W1[rb * MIDC + mm];
      const float wb0 = P.W1[ra * MIDC + 16 + mm];
      const float wb1 = P.W1[rb * MIDC + 16 + mm];
      v2f bA, bB;
      bA.x = (k0     < 17) ? wa0 : 0.f;
      bA.y = (k0 + 1 < 17) ? wa1 : 0.f;
      bB.x = (k0     < 17) ? wb0 : 0.f;
      bB.y = (k0 + 1 < 17) ? wb1 : 0.f;
      acc0 = wmma4(a1[c], bA, acc0);
      acc1 = wmma4(a1[c], bB, acc1);
    }
    ln_relu_store(L.x, acc0, acc1, P.b1, P.g1, P.be1, half, mm);

    v2f a2[8];
#pragma unroll
    for (int c = 0; c < 8; ++c) {
      const int k = 4 * c + 2 * half;
      a2[c].x = L.x[mm * 36 + k];
      a2[c].y = L.x[mm * 36 + k + 1];
    }

    // ---- layer 2: [16,32] x [32,32] ----
    acc0 = Z8; acc1 = Z8;
#pragma unroll
    for (int c = 0; c < 8; ++c) {
      const int k0 = 4 * c + 2 * half;
      v2f bA = { P.W2[k0 * MIDC + mm],      P.W2[(k0 + 1) * MIDC + mm] };
      v2f bB = { P.W2[k0 * MIDC + 16 + mm], P.W2[(k0 + 1) * MIDC + 16 + mm] };
      acc0 = wmma4(a2[c], bA, acc0);
      acc1 = wmma4(a2[c], bB, acc1);
    }
    ln_relu_store(L.x, acc0, acc1, P.b2, P.g2, P.be2, half, mm);
#pragma unroll
    for (int c = 0; c < 8; ++c) {
      const int k = 4 * c + 2 * half;
      a2[c].x = L.x[mm * 36 + k];
      a2[c].y = L.x[mm * 36 + k + 1];
    }

    // ---- layer 3 in 16-col tiles, sectioned; then per-section epilogue ----
    const int nsec = (p == 3) ? 2 : 1;
    const int sectiles = (p == 3) ? 6 : 4;
    for (int s = 0; s < nsec; ++s) {
      for (int t = 0; t < sectiles; ++t) {
        const int col0 = (s * sectiles + t) * 16;
        v8f acc = Z8;
#pragma unroll
        for (int c = 0; c < 8; ++c) {
          const int k0 = 4 * c + 2 * half;
          v2f b = { P.W3[k0 * outdim + col0 + mm],
                    P.W3[(k0 + 1) * outdim + col0 + mm] };
          acc = wmma4(a2[c], b, acc);
        }
        const float b3v = P.b3[col0 + mm];
#pragma unroll
        for (int v = 0; v < 8; ++v)
          L.R[(v + 8 * half) * 100 + t * 16 + mm] = acc[v] + b3v;
      }
      __builtin_amdgcn_wave_barrier();

      const float* Rm = &L.R[me * 100];   // R[edge][o] (+bias), o local to section
      if (p == 0) {                        // out0 += basis00 * R00 @ s0
#pragma unroll
        for (int aa = 0; aa < 4; ++aa) {
          const int a = 4 * sub + aa;
          float d = 0.f;
#pragma unroll
          for (int b = 0; b < 8; ++b) d += Rm[a * 8 + b] * s0[b];
          out0r[aa] += b00v * d;
        }
      } else if (p == 1) {                 // out1[a][j] += basis01[j]*(R01@s0)[a]
#pragma unroll
        for (int aa = 0; aa < 4; ++aa) {
          const int a = 4 * sub + aa;
          float d = 0.f;
#pragma unroll
          for (int b = 0; b < 8; ++b) d += Rm[a * 8 + b] * s0[b];
#pragma unroll
          for (int j = 0; j < 3; ++j) out1r[aa][j] += b01v[j] * d;
        }
      } else if (p == 2) {                 // out0[a] += R10 @ (basis10 . s1)
#pragma unroll
        for (int aa = 0; aa < 4; ++aa) {
          const int a = 4 * sub + aa;
          float d = 0.f;
#pragma unroll
          for (int b = 0; b < 8; ++b) d += Rm[a * 8 + b] * t10[b];
          out0r[aa] += d;
        }
      } else if (sub == s) {               // section s covers a in [4s,4s+4)
        const float* bb = &L.basis[112 + me * 27];  // basis11[j*9+i*3+f]
#pragma unroll
        for (int j = 0; j < 3; ++j)
#pragma unroll
          for (int f = 0; f < 3; ++f)
#pragma unroll
            for (int b = 0; b < 8; ++b) {
              const float vv = bb[j * 9 + f]     * s1[b][0]
                             + bb[j * 9 + 3 + f] * s1[b][1]
                             + bb[j * 9 + 6 + f] * s1[b][2];
#pragma unroll
              for (int aa = 0; aa < 4; ++aa)
                out1r[aa][j] += Rm[aa * 24 + b * 3 + f] * vv;
            }
      }
      __builtin_amdgcn_wave_barrier();
    }
  }

  // ---- write outputs: out0 [E,8] then out1 [E,24] (flat tuple) ----
#pragma unroll
  for (int aa = 0; aa < 4; ++aa) {
    const int a = 4 * sub + aa;
    out[(size_t)eg * 8 + a] = out0r[aa];
#pragma unroll
    for (int j = 0; j < 3; ++j)
      out[(size_t)E_EDGES * 8 + (size_t)eg * 24 + a * 3 + j] = out1r[aa][j];
  }
}

extern "C" void kernel_launch(void* const* d_in, const int* in_sizes, int n_in,
                              void* d_out, int out_size, void* d_ws, size_t ws_size,
                              hipStream_t stream) {
  (void)in_sizes; (void)n_in; (void)out_size; (void)d_ws; (void)ws_size;
  const float* h0  = (const float*)d_in[0];
  const float* h1  = (const float*)d_in[1];
  /* d_in[2] = positions (unused by reference) */
  const float* rr  = (const float*)d_in[3];
  const float* ef  = (const float*)d_in[4];
  const int*   ei  = (const int*)d_in[5];     // row = edge_index[0] = first E ints
  const float* b00 = (const float*)d_in[6];
  const float* b01 = (const float*)d_in[7];
  const float* b10 = (const float*)d_in[8];
  const float* b11 = (const float*)d_in[9];

  AllParams ap;
  for (int p = 0; p < 4; ++p) {
    const int base = 10 + 10 * p;   // params['00','01','10','11'] x {W1,b1,g1,be1,W2,b2,g2,be2,W3,b3}
    ap.p[p].W1  = (const float*)d_in[base + 0];
    ap.p[p].b1  = (const float*)d_in[base + 1];
    ap.p[p].g1  = (const float*)d_in[base + 2];
    ap.p[p].be1 = (const float*)d_in[base + 3];
    ap.p[p].W2  = (const float*)d_in[base + 4];
    ap.p[p].b2  = (const float*)d_in[base + 5];
    ap.p[p].g2  = (const float*)d_in[base + 6];
    ap.p[p].be2 = (const float*)d_in[base + 7];
    ap.p[p].W3  = (const float*)d_in[base + 8];
    ap.p[p].b3  = (const float*)d_in[base + 9];
  }

  const dim3 grid(E_EDGES / (16 * WAVES));   // 2500 blocks, exact
  const dim3 block(32 * WAVES);              // 4 waves (wave32) per block
  se3_partial_wmma<<<grid, block, 0, stream>>>(h0, h1, rr, ef, ei,
                                               b00, b01, b10, b11, ap,
                                               (float*)d_out);
}